// LIFNeuron_49374944034992
// MI455X (gfx1250) — compile-verified
//
#include <hip/hip_runtime.h>
#include <cstdint>

// LIF: V = 0.9*V + I; spike = (V >= 1); V -= spike.
// current [8, 4096, 1024] f32 -> spikes [8, 4096, 1024] f32 ++ mem_final [8, 1024] f32.

constexpr int B = 8;
constexpr int S = 4096;
constexpr int D = 1024;

constexpr int DBLK  = 32;          // chains per wave (one per lane)
constexpr int TSTEP = 64;          // timesteps per TDM tile
constexpr int NBUF  = 12;          // TDM pipeline depth (12 x 8KB = 96KB LDS, 24MB in flight fleet-wide)
constexpr int NCHUNK = S / TSTEP;  // 64

typedef unsigned int u32x4 __attribute__((ext_vector_type(4)));
typedef int          i32x8 __attribute__((ext_vector_type(8)));
typedef int          i32x4 __attribute__((ext_vector_type(4)));

__device__ __forceinline__ unsigned rfl(unsigned v) {
    return (unsigned)__builtin_amdgcn_readfirstlane((int)v);
}

// Issue one TDM load: 2D tile [TSTEP rows x DBLK floats], row stride D floats,
// from global address `gsrc` into LDS byte offset `lds_off`.
__device__ __forceinline__ void tdm_load_tile(const float* gsrc, unsigned lds_off) {
    const uint64_t ga = (uint64_t)(uintptr_t)gsrc;

    u32x4 g0;
    g0.x = rfl(1u);                                    // count=1, user mode, no gather
    g0.y = rfl(lds_off);                               // lds_addr (bytes)
    g0.z = rfl((unsigned)(ga & 0xFFFFFFFFull));        // global_addr[31:0]
    g0.w = rfl((unsigned)((ga >> 32) & 0x1FFFFFFull)   // global_addr[56:32]
               | (2u << 30));                          // type = 2 ("image")

    const uint64_t dim1_stride = (uint64_t)S * (uint64_t)D;   // elements
    i32x8 g1;
    g1[0] = (int)rfl(2u << 16);                        // wg_mask=0, data_size=2 (4B)
    g1[1] = (int)rfl(((unsigned)D & 0xFFFFu) << 16);   // tensor_dim0 lo16 = 1024
    g1[2] = (int)rfl(((unsigned)D >> 16)               // tensor_dim0 hi16 = 0
               | (((unsigned)S & 0xFFFFu) << 16));     // tensor_dim1 lo16 = 4096
    g1[3] = (int)rfl(((unsigned)S >> 16)               // tensor_dim1 hi16 = 0
               | ((unsigned)DBLK << 16));              // tile_dim0 = 32
    g1[4] = (int)rfl((unsigned)TSTEP);                 // tile_dim1 = 64, tile_dim2 = 0
    g1[5] = (int)rfl((unsigned)D);                     // tensor_dim0_stride lo32 = 1024
    g1[6] = (int)rfl((unsigned)(((dim1_stride) & 0xFFFFull) << 16)); // stride0 hi16=0 | stride1 lo16
    g1[7] = (int)rfl((unsigned)(dim1_stride >> 16));   // tensor_dim1_stride[47:16]

    const i32x4 z4 = {0, 0, 0, 0};                     // groups 2/3 unused (2D tensor)
    const i32x8 z8 = {0, 0, 0, 0, 0, 0, 0, 0};
    __builtin_amdgcn_tensor_load_to_lds(g0, g1, z4, z4, z8, 0);
}

__global__ void __launch_bounds__(DBLK)
lif_tdm_kernel(const float* __restrict__ cur,
               const float* __restrict__ mem0,
               float* __restrict__ spikes,
               float* __restrict__ memf) {
    __shared__ float smem[NBUF * TSTEP * DBLK];        // 96 KB (of 320 KB/WGP)

    const int lane = threadIdx.x;                      // 0..31
    const int blk  = blockIdx.x;                       // 0..255
    const int b    = blk / (D / DBLK);                 // 0..7
    const int dblk = blk % (D / DBLK);                 // 0..31
    const int d    = dblk * DBLK + lane;

    const float* curBase = cur    + ((size_t)b * S) * D + (size_t)dblk * DBLK;
    float*       spkBase = spikes + ((size_t)b * S) * D + (size_t)dblk * DBLK;

    // Low 32 bits of the flat address of an LDS object == byte offset in LDS.
    const unsigned ldsBase = (unsigned)(uintptr_t)(&smem[0]);
    constexpr unsigned BUF_BYTES = TSTEP * DBLK * 4;   // 8 KB

    // Prologue: fill the pipeline with NBUF outstanding tensor loads.
    #pragma unroll
    for (int c = 0; c < NBUF; ++c) {
        tdm_load_tile(curBase + (size_t)c * TSTEP * D, ldsBase + (unsigned)c * BUF_BYTES);
    }

    float mem = mem0[(size_t)b * D + d];

    for (int c = 0; c < NCHUNK; ++c) {
        const int bi = c % NBUF;
        const float* lbuf = &smem[(size_t)bi * TSTEP * DBLK];

        // TDM ops complete in order per wave: chunk c is done once <= NBUF-1 remain.
        __builtin_amdgcn_s_wait_tensorcnt((short)(NBUF - 1));
        asm volatile("" ::: "memory");                 // LDS now holds chunk c

        float* so = spkBase + (size_t)c * TSTEP * D;
        #pragma unroll
        for (int t = 0; t < TSTEP; ++t) {
            const float I = lbuf[t * DBLK + lane];     // ds_load_b32, bank-conflict-free
            mem = __builtin_fmaf(0.9f, mem, I);        // integrate with leak
            // Critical path: fma -> cmp -> cndmask -> fma (3 deep).
            // msub and the spike-value select sit off the chain.
            const bool  sp   = (mem >= 1.0f);
            const float msub = mem - 1.0f;             // independent of the compare
            so[(size_t)t * D + lane] = sp ? 1.0f : 0.0f;
            mem = sp ? msub : mem;                     // soft reset (theta = 1.0)
        }

        // All ds reads of this buffer must land before the TDM overwrites it.
        asm volatile("s_wait_dscnt 0" ::: "memory");

        const int cn = c + NBUF;
        if (cn < NCHUNK) {
            tdm_load_tile(curBase + (size_t)cn * TSTEP * D,
                          ldsBase + (unsigned)bi * BUF_BYTES);
        }
    }

    memf[(size_t)b * D + d] = mem;
}

extern "C" void kernel_launch(void* const* d_in, const int* in_sizes, int n_in,
                              void* d_out, int out_size, void* d_ws, size_t ws_size,
                              hipStream_t stream) {
    const float* cur  = (const float*)d_in[0];   // [8, 4096, 1024]
    const float* mem0 = (const float*)d_in[1];   // [8, 1024]
    float* spikes = (float*)d_out;                       // [8, 4096, 1024]
    float* memf   = (float*)d_out + (size_t)B * S * D;   // [8, 1024]

    dim3 grid(B * (D / DBLK));   // 256 single-wave workgroups
    dim3 block(DBLK);            // 32 threads = 1 wave32
    hipLaunchKernelGGL(lif_tdm_kernel, grid, block, 0, stream, cur, mem0, spikes, memf);
}